// DynamicGraphLearner_19069654794980
// MI455X (gfx1250) — compile-verified
//
#include <hip/hip_runtime.h>
#include <hip/hip_bf16.h>
#include <math.h>

// ---------------------------------------------------------------------------
// DynamicGraphLearner for MI455X (gfx1250, wave32, WMMA)
//
// Pipeline:
//   k1: w_comb[16][256] = fold att into W (rows 8..15 zero-padded for WMMA B)
//   k2: s[N][8]         = emb @ w_comb^T  via V_WMMA_F32_16X16X4_F32
//   k3: corr dense NxN  = zero + edge scatter (in d_ws)
//   k4: per-row fused score + top-15 + softmax -> edge_index/edge_weight
// ---------------------------------------------------------------------------

#define N_NODES   4096
#define HDIM      256
#define NHEADS    4
#define HEADD     64
#define TOPK      15
#define NEG_SLOPE 0.2f

typedef __attribute__((ext_vector_type(2))) float v2f;
typedef __attribute__((ext_vector_type(8))) float v8f;

// ---------------- k1: fold attention vectors into W (16 rows, 8..15 = 0) ----
// w_comb[j][k] = sum_d W[(j&3)*64+d][k] * att[j&3][ (j>>2)*64 + d ],  j in 0..7
// w_comb[j][k] = 0 for j in 8..15  (zero pad so WMMA B needs no EXEC masking)
__global__ __launch_bounds__(256) void build_wcomb(const float* __restrict__ W,
                                                   const float* __restrict__ att,
                                                   float* __restrict__ wcomb) {
    const int j = blockIdx.x;        // 0..15
    const int k = threadIdx.x;       // 0..255
    float acc = 0.0f;
    if (j < 8) {
        const int head = j & 3;
        const int isSrc = j >> 2;
        #pragma unroll 8
        for (int d = 0; d < HEADD; ++d) {
            acc += W[(head * HEADD + d) * HDIM + k] * att[head * (2 * HEADD) + isSrc * HEADD + d];
        }
    }
    wcomb[j * HDIM + k] = acc;
}

// ---------------- k2: s = emb @ wcomb^T via f32 WMMA ----------------
// One wave (32 lanes) per 16-row tile. B is 16x256 (rows 8..15 zero), so all
// loads are unconditional -> no saveexec divergence in the hot loop.
__global__ __launch_bounds__(32) void s_gemm_wmma(const float* __restrict__ emb,
                                                  const float* __restrict__ wcomb,
                                                  float* __restrict__ s) {
    const int tile = blockIdx.x;          // 256 tiles of 16 rows
    const int lane = threadIdx.x;         // 0..31
    const int half = lane >> 4;           // 0: lanes 0-15, 1: lanes 16-31
    const int l15  = lane & 15;

    const float* __restrict__ arow = emb   + (size_t)(tile * 16 + l15) * HDIM + half * 2;
    const float* __restrict__ brow = wcomb + (size_t)l15 * HDIM + half * 2;

    v8f acc = {};
    #pragma unroll 4
    for (int k = 0; k < HDIM; k += 4) {
        v2f a, b;
        a.x = arow[k];      // A: (M=l15, K=k+half*2)
        a.y = arow[k + 1];
        b.x = brow[k];      // B: (K=k+half*2, N=l15), rows>=8 are zeros
        b.y = brow[k + 1];
        // 8 args: (neg_a, A, neg_b, B, c_mod, C, reuse_a, reuse_b)
        acc = __builtin_amdgcn_wmma_f32_16x16x4_f32(false, a, false, b,
                                                    (short)0, acc, false, false);
    }

    // D layout: VGPR r -> (M=r, N=l15) for lanes 0-15, (M=r+8, N=l15) for 16-31
    if (l15 < 8) {
        #pragma unroll
        for (int r = 0; r < 8; ++r) {
            const int m = r + half * 8;
            s[(size_t)(tile * 16 + m) * 8 + l15] = acc[r];
        }
    }
}

// ---------------- k3: dense corr scratch ----------------
__global__ __launch_bounds__(256) void zero_corr(float4* __restrict__ corr4, int n4) {
    int i = blockIdx.x * blockDim.x + threadIdx.x;
    const int stride = gridDim.x * blockDim.x;
    const float4 z = make_float4(0.f, 0.f, 0.f, 0.f);
    for (; i < n4; i += stride) corr4[i] = z;
}

__global__ __launch_bounds__(256) void scatter_corr(const int* __restrict__ eidx,
                                                    const float* __restrict__ ew,
                                                    float* __restrict__ corr, int ec) {
    const int e = blockIdx.x * blockDim.x + threadIdx.x;
    if (e >= ec) return;
    const int r = eidx[e];          // corr_edge_index[0][e]
    const int c = eidx[ec + e];     // corr_edge_index[1][e]
    corr[(size_t)r * N_NODES + c] = ew[e];   // last-write-wins, matches scatter-set
}

// ---------------- k4: fused score + top-k + softmax ----------------
__global__ __launch_bounds__(256) void row_topk(const float* __restrict__ s,
                                                const unsigned char* __restrict__ secmask,
                                                const unsigned char* __restrict__ act,
                                                const float* __restrict__ corr,
                                                const float* __restrict__ lamp,
                                                int* __restrict__ out_idx,
                                                float* __restrict__ out_w) {
    const int i   = blockIdx.x;          // dst row
    const int tid = threadIdx.x;         // 0..255
    const float lam = lamp[0];

    // row-constant dst scores
    const float4 sd = *(const float4*)(s + (size_t)i * 8);
    const bool act_i = act[i] != 0;

    const float*         crow = corr    + (size_t)i * N_NODES;
    const unsigned char* mrow = secmask + (size_t)i * N_NODES;

    // local sorted top-15; each thread scans 16 columns as 4 groups of 4
    float lv[TOPK]; int li[TOPK];
    #pragma unroll
    for (int t = 0; t < TOPK; ++t) { lv[t] = -INFINITY; li[t] = 0x7fffffff; }

    for (int m = 0; m < N_NODES / 1024; ++m) {
        const int j0 = tid * 4 + m * 1024;                 // 4 consecutive columns
        const float4 c4 = *(const float4*)(crow + j0);     // 16B corr load
        const uchar4 m4 = *(const uchar4*)(mrow + j0);     // 4B mask load
        const unsigned char mk[4] = { m4.x, m4.y, m4.z, m4.w };
        const float cv[4] = { c4.x, c4.y, c4.z, c4.w };

        #pragma unroll
        for (int q = 0; q < 4; ++q) {
            const int j = j0 + q;
            const float4 ss = *(const float4*)(s + (size_t)j * 8 + 4);
            float a0 = sd.x + ss.x, a1 = sd.y + ss.y, a2 = sd.z + ss.z, a3 = sd.w + ss.w;
            a0 = (a0 > 0.f) ? a0 : NEG_SLOPE * a0;
            a1 = (a1 > 0.f) ? a1 : NEG_SLOPE * a1;
            a2 = (a2 > 0.f) ? a2 : NEG_SLOPE * a2;
            a3 = (a3 > 0.f) ? a3 : NEG_SLOPE * a3;
            const float base = (a0 + a1 + a2 + a3) * 0.25f;   // mean over heads, T=1
            const bool msk = act_i && (act[j] != 0) && (mk[q] != 0);
            const float sc = msk ? (base + lam * cv[q]) : -INFINITY;

            // stable insertion (lower index wins ties; j ascending within thread)
            if (sc > lv[TOPK - 1] || (sc == lv[TOPK - 1] && j < li[TOPK - 1])) {
                int p = TOPK - 1;
                while (p > 0 && (lv[p - 1] < sc || (lv[p - 1] == sc && li[p - 1] > j))) {
                    lv[p] = lv[p - 1]; li[p] = li[p - 1]; --p;
                }
                lv[p] = sc; li[p] = j;
            }
        }
    }

    // block-wide 15-round argmax extraction
    __shared__ float sv[256];
    __shared__ int   si[256];
    __shared__ int   st[256];

    float topv[TOPK]; int topi[TOPK];
    int ptr = 0;

    for (int r = 0; r < TOPK; ++r) {
        sv[tid] = (ptr < TOPK) ? lv[ptr] : -INFINITY;
        si[tid] = (ptr < TOPK) ? li[ptr] : 0x7fffffff;
        st[tid] = tid;
        __syncthreads();
        for (int off = 128; off > 0; off >>= 1) {
            if (tid < off) {
                const float v2 = sv[tid + off]; const int i2 = si[tid + off];
                if (v2 > sv[tid] || (v2 == sv[tid] && i2 < si[tid])) {
                    sv[tid] = v2; si[tid] = i2; st[tid] = st[tid + off];
                }
            }
            __syncthreads();
        }
        topv[r] = sv[0]; topi[r] = si[0];
        if (tid == st[0]) ++ptr;
        __syncthreads();
    }

    // softmax over the 15 winners; threads 0..14 write output
    if (tid < TOPK) {
        const float mx = topv[0];
        float sum = 0.f;
        #pragma unroll
        for (int r = 0; r < TOPK; ++r) sum += __expf(topv[r] - mx);
        const float w = __expf(topv[tid] - mx) / sum;
        const int base = i * TOPK + tid;
        const int NK = N_NODES * TOPK;
        out_idx[base]      = topi[tid];   // edge_index[0] = topk column (source)
        out_idx[NK + base] = i;           // edge_index[1] = row (dest)
        out_w[base] = w;
    }
}

// ---------------------------------------------------------------------------
extern "C" void kernel_launch(void* const* d_in, const int* in_sizes, int n_in,
                              void* d_out, int out_size, void* d_ws, size_t ws_size,
                              hipStream_t stream) {
    const float*         emb    = (const float*)d_in[0];          // (4096,256)
    const float*         W      = (const float*)d_in[1];          // (256,256)
    const float*         att    = (const float*)d_in[2];          // (4,128)
    const float*         lam    = (const float*)d_in[3];          // (1,)
    const unsigned char* secm   = (const unsigned char*)d_in[4];  // (4096,4096) bool
    const unsigned char* actm   = (const unsigned char*)d_in[5];  // (4096,) bool
    const int*           eidx   = (const int*)d_in[6];            // (2,E_c)
    const float*         ew     = (const float*)d_in[7];          // (E_c,)
    const int            ec     = in_sizes[7];

    // workspace layout (floats): [wcomb 16*256][s 4096*8][corr 4096*4096]
    float* ws     = (float*)d_ws;
    float* wcomb  = ws;                                   // 4096 floats
    float* s      = ws + 16 * HDIM;                       // offset 4096
    float* corr   = ws + 16 * HDIM + N_NODES * 8;         // offset 36864 (16B aligned)

    const int NK = N_NODES * TOPK;
    int*   out_idx = (int*)d_out;                // first 2*NK elements: edge_index
    float* out_w   = (float*)d_out + 2 * NK;     // last NK elements: edge_weight

    build_wcomb<<<16, 256, 0, stream>>>(W, att, wcomb);

    s_gemm_wmma<<<N_NODES / 16, 32, 0, stream>>>(emb, wcomb, s);

    const int n4 = (N_NODES * N_NODES) / 4;
    zero_corr<<<4096, 256, 0, stream>>>((float4*)corr, n4);
    scatter_corr<<<(ec + 255) / 256, 256, 0, stream>>>(eidx, ew, corr, ec);

    row_topk<<<N_NODES, 256, 0, stream>>>(s, secm, actm, corr, lam, out_idx, out_w);
}